// MessagePassing_49074296324425
// MI455X (gfx1250) — compile-verified
//
#include <hip/hip_runtime.h>
#include <stdint.h>

// Problem constants (fixed by reference):
#define NB      8
#define CB      64
#define HH      128
#define WW      128
#define TAPS    9
#define NSTEPS  10
#define EPSV    1e-5f

// Tiling:
#define TILE    32
#define HALO    10
#define IN      52                  // TILE + 2*HALO
#define AREA    (IN*IN)             // 2704
#define PITCH   64                  // LDS row pitch (floats) == TDM tile_dim0
#define BROWS   54                  // IN + top/bottom zero rows
#define BUF     (BROWS*PITCH)       // 3456 floats / buffer
#define CPB     8                   // channels per block
#define NTHR    256
#define NQUADS  (IN*(IN/4))         // 676
#define SMEM_FLOATS (TAPS*AREA + 3*BUF)
#define SMEM_BYTES  (SMEM_FLOATS*4) // 138,816 B

#if __has_builtin(__builtin_amdgcn_global_load_async_to_lds_b32)
#define USE_ASYNC 1
#pragma message("gfx1250 async-to-LDS path: ENABLED")
#else
#define USE_ASYNC 0
#pragma message("gfx1250 async-to-LDS path: DISABLED (builtin not found)")
#endif

#if __has_builtin(__builtin_amdgcn_tensor_load_to_lds)
#define USE_TDM 1
#if __clang_major__ >= 23
#define TDM_ARITY6 1
#pragma message("gfx1250 TDM path: ENABLED (6-arg builtin)")
#else
#define TDM_ARITY6 0
#pragma message("gfx1250 TDM path: ENABLED (5-arg builtin)")
#endif
#else
#define USE_TDM 0
#pragma message("gfx1250 TDM path: DISABLED (builtin not found)")
#endif

typedef __attribute__((address_space(3))) int lds_int_t;   // LDS ("__shared__") int
typedef __attribute__((address_space(1))) int gbl_int_t;   // global ("__device__") int
typedef __attribute__((ext_vector_type(4))) unsigned int u32x4;
typedef __attribute__((ext_vector_type(8))) int i32x8;
typedef __attribute__((ext_vector_type(4))) int i32x4;

__device__ __forceinline__ lds_int_t* to_lds(void* p) {
  // Generic LDS address -> 32-bit LDS offset (addr[31:0] per CDNA5 aperture rules)
  return (lds_int_t*)(uint32_t)(uintptr_t)p;
}
__device__ __forceinline__ gbl_int_t* to_gbl(const void* p) {
  return (gbl_int_t*)(uintptr_t)p;
}

__device__ __forceinline__ void wait_async0() {
#if __has_builtin(__builtin_amdgcn_s_wait_asynccnt)
  __builtin_amdgcn_s_wait_asynccnt(0);
#else
  asm volatile("s_wait_asynccnt 0x0" ::: "memory");
#endif
}

__device__ __forceinline__ void wait_tensor0() {
#if __has_builtin(__builtin_amdgcn_s_wait_tensorcnt)
  __builtin_amdgcn_s_wait_tensorcnt(0);
#else
  asm volatile("s_wait_tensorcnt 0x0" ::: "memory");
#endif
}

#if USE_TDM
// One-instruction DMA of a 64x52 fp32 tile (row stride 128 floats in memory)
// into LDS at pitch 64 floats. rem_x/rem_y = distance from tile origin to the
// tensor edge in elements; TDM zero-fills reads beyond those (right/bottom OOB).
__device__ __forceinline__ void tdm_issue(const float* src_tile, void* lds_dst,
                                          int rem_x, int rem_y) {
  const uint64_t ga = (uint64_t)(uintptr_t)src_tile;
  const uint32_t la = (uint32_t)(uintptr_t)lds_dst;
  u32x4 g0;
  g0.x = 1u;                                   // count=1, is_restore=0, gather=0
  g0.y = la;                                   // lds_addr (bytes)
  g0.z = (uint32_t)ga;                         // global_addr[31:0]
  g0.w = (uint32_t)(ga >> 32) | (2u << 30);    // global_addr[56:32] | type=2
  i32x8 g1;
  g1[0] = (2 << 16);                           // data_size=4B; mask/pad/iter=0
  g1[1] = (rem_x & 0xFFFF) << 16;              // tensor_dim0[15:0] @ bits[63:48]
  g1[2] = (int)(((unsigned)rem_x >> 16) | (((unsigned)rem_y & 0xFFFFu) << 16));
  g1[3] = (int)(((unsigned)rem_y >> 16) | (64u << 16));  // tile_dim0=64
  g1[4] = IN;                                  // tile_dim1=52 rows, tile_dim2=0
  g1[5] = WW;                                  // tensor_dim0_stride = 128
  g1[6] = 0;
  g1[7] = 0;
  const i32x4 z4 = {0, 0, 0, 0};
#if TDM_ARITY6
  const i32x8 z8 = {0, 0, 0, 0, 0, 0, 0, 0};
  __builtin_amdgcn_tensor_load_to_lds(g0, g1, z4, z4, z8, 0);
#else
  __builtin_amdgcn_tensor_load_to_lds(g0, g1, z4, z4, 0);
#endif
}
#endif

__device__ __forceinline__ float4 ld4(const float* p) {
  return *(const float4*)p;
}

__global__ __launch_bounds__(NTHR)
void mp_fused(const float* __restrict__ gin,
              const float* __restrict__ gw,
              float* __restrict__ gout) {
  extern __shared__ float smem[];
  float* wn = smem;                 // [TAPS][AREA] normalized weights
  float* xb = smem + TAPS*AREA;     // 3 buffers of BUF floats each

  const int tid = threadIdx.x;
  const int tx  = blockIdx.x & 3;
  const int ty  = blockIdx.x >> 2;
  const int cg  = blockIdx.y;       // channel group
  const int nb  = blockIdx.z;       // batch
  const int gy0 = ty*TILE - HALO;
  const int gx0 = tx*TILE - HALO;
#if USE_TDM
  const bool useTDM = (tx > 0) && (ty > 0);   // origin >= 0: TDM-safe rectangle
#else
  const bool useTDM = false;
#endif

  // ---- zero all LDS x-buffers (ring + OOB invariant) ----
  for (int i = tid; i < 3*BUF; i += NTHR) xb[i] = 0.0f;

  // ---- Phase 1: per-pixel weight normalization (shared across channels) ----
  const float* wbase = gw + (size_t)nb * TAPS * HH * WW;
  for (int p = tid; p < AREA; p += NTHR) {
    const int ly = p / IN, lx = p - ly*IN;
    const int gy = gy0 + ly, gx = gx0 + lx;
    if (gy >= 0 && gy < HH && gx >= 0 && gx < WW) {
      const float* wp = wbase + gy*WW + gx;
      float tw[TAPS]; float s = 0.0f;
#pragma unroll
      for (int t = 0; t < TAPS; ++t) { tw[t] = wp[t*(HH*WW)]; s += tw[t]; }
      const float inv = 1.0f / (s + EPSV);
#pragma unroll
      for (int t = 0; t < TAPS; ++t) wn[t*AREA + p] = tw[t]*inv;
    } else {
#pragma unroll
      for (int t = 0; t < TAPS; ++t) wn[t*AREA + p] = 0.0f;
    }
  }
  __syncthreads();  // zero-init + weights visible before any DMA write lands

  // ---- channel-tile loader: TDM for interior-origin tiles, async per-lane else
  auto issue_load = [&](int bi, int gc) {
    const float* src = gin + ((size_t)(nb*CB + gc)) * HH * WW;
    float* dst = xb + bi*BUF;
#if USE_TDM
    if (useTDM) {
      if (tid < 32) {   // one wave issues the wave-level TDM op
        tdm_issue(src + gy0*WW + gx0, dst + PITCH + 4, WW - gx0, HH - gy0);
      }
      return;
    }
#endif
    for (int p = tid; p < AREA; p += NTHR) {
      const int ly = p / IN, lx = p - ly*IN;
      const int gy = gy0 + ly, gx = gx0 + lx;
      if (gy >= 0 && gy < HH && gx >= 0 && gx < WW) {
        float* lp = dst + (ly+1)*PITCH + (lx+4);
#if USE_ASYNC
        __builtin_amdgcn_global_load_async_to_lds_b32(
            to_gbl(src + gy*WW + gx), to_lds(lp), 0, 0);
#else
        *lp = src[gy*WW + gx];
#endif
      }
    }
  };

  int cur = 0;
  issue_load(cur, cg*CPB + 0);

  for (int c = 0; c < CPB; ++c) {
    if (useTDM) {
      if (tid < 32) wait_tensor0();
    } else {
      wait_async0();
    }
    __syncthreads();                       // tile for channel c resident
    if (c + 1 < CPB) issue_load(1 - cur, cg*CPB + c + 1);  // overlap with compute

    float* A = xb + cur*BUF;               // current state
    float* B = xb + 2*BUF;                 // scratch

#pragma unroll 1
    for (int s = 0; s < NSTEPS; ++s) {
      for (int q = tid; q < NQUADS; q += NTHR) {
        const int y  = q / 13;
        const int qx = q - y*13;
        const int cb = 4*qx;               // 16B-aligned buffer column base
        const float* w0 = wn + y*IN + cb;
        float accx = 0.f, accy = 0.f, accz = 0.f, accw = 0.f;
#pragma unroll
        for (int di = 0; di < 3; ++di) {
          const float* r = A + (y + di)*PITCH + cb;
          const float4 a4 = ld4(r), b4 = ld4(r + 4), c4 = ld4(r + 8);
          const float v[12] = { a4.x, a4.y, a4.z, a4.w,
                                b4.x, b4.y, b4.z, b4.w,
                                c4.x, c4.y, c4.z, c4.w };
#pragma unroll
          for (int dj = 0; dj < 3; ++dj) {
            const float4 wv = ld4(w0 + (di*3 + dj)*AREA);
            accx = fmaf(wv.x, v[3 + dj], accx);
            accy = fmaf(wv.y, v[4 + dj], accy);
            accz = fmaf(wv.z, v[5 + dj], accz);
            accw = fmaf(wv.w, v[6 + dj], accw);
          }
        }
        float4 o; o.x = accx; o.y = accy; o.z = accz; o.w = accw;
        *(float4*)(B + (y+1)*PITCH + cb + 4) = o;
      }
      float* t = A; A = B; B = t;          // even #steps -> result back in xb[cur]
      __syncthreads();
    }

    // ---- store the clean 32x32 core ----
    const int gc = cg*CPB + c;
    const size_t ob = ((size_t)(nb*CB + gc)) * HH * WW;
    for (int o = tid; o < TILE*TILE; o += NTHR) {
      const int oy = o >> 5, ox = o & 31;
      gout[ob + (size_t)(ty*TILE + oy)*WW + (tx*TILE + ox)] =
          A[(oy + HALO + 1)*PITCH + (ox + HALO + 4)];
    }
    cur = 1 - cur;
    __syncthreads();
  }
}

extern "C" void kernel_launch(void* const* d_in, const int* in_sizes, int n_in,
                              void* d_out, int out_size, void* d_ws, size_t ws_size,
                              hipStream_t stream) {
  (void)in_sizes; (void)n_in; (void)out_size; (void)d_ws; (void)ws_size;
  const float* gin = (const float*)d_in[0];   // [8,64,128,128] fp32
  const float* gw  = (const float*)d_in[1];   // [8,9,128,128]  fp32
  float* gout = (float*)d_out;                // [8,64,128,128] fp32

  (void)hipFuncSetAttribute((const void*)mp_fused,
                            hipFuncAttributeMaxDynamicSharedMemorySize,
                            SMEM_BYTES);

  dim3 grid(16 /*spatial tiles*/, CB/CPB /*channel groups*/, NB /*batch*/);
  mp_fused<<<grid, NTHR, SMEM_BYTES, stream>>>(gin, gw, gout);
}